// customLoss_48773648614415
// MI455X (gfx1250) — compile-verified
//
#include <hip/hip_runtime.h>
#include <math.h>

// CDNA5 / gfx1250, wave32.
typedef __attribute__((ext_vector_type(2))) float v2f;
typedef __attribute__((ext_vector_type(4))) float f4;
typedef __attribute__((ext_vector_type(8))) float v8f;

#define LOG_SQRT_2PI 0.9189385332046727f

// ---------------------------------------------------------------------------
// Kernel 1: per-row Gaussian-NLL prior partial sums (tiny: B x 7).
// 64 blocks x 128 threads = 8192 threads, one row each. Deterministic
// fixed-order block reduction -> ws[blockIdx.x].
// ---------------------------------------------------------------------------
__global__ void __launch_bounds__(128) prior_partial_kernel(
    const float* __restrict__ pred, const float* __restrict__ sigma,
    float* __restrict__ partials, int nrows) {
  const int row = blockIdx.x * 128 + threadIdx.x;
  float acc = 0.0f;
  if (row < nrows) {
    float p[7], s[7];
#pragma unroll
    for (int i = 0; i < 7; ++i) {
      p[i] = pred[row * 7 + i];
      s[i] = sigma[row * 7 + i];
    }
    const float O2 = p[0], N2 = p[1], H2 = p[2], CO2 = p[3], H2O = p[4],
                CH4 = p[5], NH3 = p[6];
    const float H = 2.f * H2 + 2.f * H2O + 3.f * NH3 + 4.f * CH4;
    const float C = CO2 + CH4;
    const float O = 2.f * O2 + 2.f * CO2 + H2O;
    const float N = 2.f * N2 + NH3;

    float mu[7] = {0.f, 0.f, 0.f, 0.f, 0.f, 0.f, 0.f};
    if (H > 2.f * O + 4.f * C) {
      if (3.f * N < H - 2.f * O - 4.f * C) {  // branch A1
        const float D = H - N - 2.f * C;
        mu[2] = (H - 2.f * O - 4.f * C - 3.f * N) / D;
        mu[4] = 2.f * O / D;
        mu[5] = 2.f * C / D;
        mu[6] = 2.f * N / D;
      } else {  // branch A2
        const float D = H + 2.f * C + 3.f * N + 4.f * O;
        mu[1] = (3.f * N + 4.f * C + 2.f * O - H) / D;
        mu[4] = 6.f * O / D;
        mu[5] = 6.f * C / D;
        mu[6] = (2.f * H - 8.f * C - 4.f * O) / D;
      }
    } else if (2.f * O > H + 4.f * C) {  // branch B
      const float D = H + 2.f * O + 2.f * N;
      mu[0] = (2.f * O - H - 4.f * C) / D;
      mu[1] = 2.f * N / D;
      mu[3] = 4.f * C / D;
      mu[4] = 2.f * H / D;
    } else if (fabsf(H + C + O + N - 1.0f) < 1e-3f) {  // branch C
      const float D1 = H + 2.f * O + 2.f * N;
      const float D2 = 2.f * H + 4.f * O + 4.f * N;
      mu[1] = 2.f * N / D1;
      mu[3] = (2.f * O + 4.f * C - H) / D2;
      mu[4] = (H + 2.f * O - 4.f * C) / D1;
      mu[5] = (H - 2.f * O + 4.f * C) / D2;
    }
#pragma unroll
    for (int i = 0; i < 7; ++i) {
      const float d = p[i] - mu[i];
      acc += LOG_SQRT_2PI + logf(s[i]) + d * d / (2.f * s[i] * s[i]);
    }
  }
  __shared__ float red[128];
  red[threadIdx.x] = acc;
  __syncthreads();
#pragma unroll
  for (int stride = 64; stride > 0; stride >>= 1) {
    if (threadIdx.x < stride) red[threadIdx.x] += red[threadIdx.x + stride];
    __syncthreads();
  }
  if (threadIdx.x == 0) partials[blockIdx.x] = red[0];
}

// ---------------------------------------------------------------------------
// Kernel 2: fold 64 partials -> ws[64] = prior scalar (sum over [B,7] / B).
// ---------------------------------------------------------------------------
__global__ void prior_finalize_kernel(float* __restrict__ ws, int nrows) {
  __shared__ float red[64];
  const int t = threadIdx.x;
  red[t] = ws[t];
  __syncthreads();
#pragma unroll
  for (int stride = 32; stride > 0; stride >>= 1) {
    if (t < stride) red[t] += red[t + stride];
    __syncthreads();
  }
  if (t == 0) ws[64] = red[0] / (float)nrows;
}

// ---------------------------------------------------------------------------
// Kernel 3: bandwidth-bound likelihood + posterior.
// Each block owns a 16-row tile of the [B, S] spectra. 16 threads/row stream
// the row with coalesced 128-bit non-temporal loads (data is single-use;
// 256 MB > 192 MB L2). 16x16 partial-sum matrix P lands in LDS; wave 0 then
// computes all 16 row sums as P x ones via 4 chained V_WMMA_F32_16X16X4_F32.
// ---------------------------------------------------------------------------
__global__ void __launch_bounds__(256) posterior_kernel(
    const float* __restrict__ yReal, const float* __restrict__ ySim,
    const float* __restrict__ ws, float* __restrict__ out, int S) {
  __shared__ float P[256];  // P[row*16 + sub]

  const int t = threadIdx.x;
  const int rowInTile = t >> 4;  // 0..15
  const int sub = t & 15;        // 16 threads per row
  const long row = (long)blockIdx.x * 16 + rowInTile;
  const long rowOff = row * (long)S;

  const f4* __restrict__ yr = (const f4*)(yReal + rowOff) + sub;
  const f4* __restrict__ ys = (const f4*)(ySim + rowOff) + sub;

  // Each iteration: 16 threads x 16B = 256B contiguous per row.
  const int iters = S >> 6;  // S / (16 threads * 4 floats)
  float acc = 0.0f;
#pragma unroll 4
  for (int i = 0; i < iters; ++i) {
    const f4 a = __builtin_nontemporal_load(yr + (long)i * 16);
    const f4 b = __builtin_nontemporal_load(ys + (long)i * 16);
    const float dx = a.x - b.x, dy = a.y - b.y, dz = a.z - b.z,
                dw = a.w - b.w;
    acc += dx * dx + dy * dy + dz * dz + dw * dw;
  }
  P[t] = acc;
  __syncthreads();

  if (t < 32) {  // wave 0 only; uniform branch -> EXEC all-ones for WMMA
    const int half = t >> 4;  // 0: K0/K1 lanes, 1: K2/K3 lanes
    const int m = t & 15;     // matrix row this lane feeds
    const float* pr = &P[m * 16];

    v2f bone;
    bone[0] = 1.0f;
    bone[1] = 1.0f;  // B = ones(4x16): layout-invariant
    v8f c = {};
#pragma unroll
    for (int kk = 0; kk < 4; ++kk) {
      // A-layout (f32 16x4): lanes 0-15 hold K=0,1; lanes 16-31 hold K=2,3.
      v2f a;
      a[0] = pr[kk * 4 + half * 2 + 0];
      a[1] = pr[kk * 4 + half * 2 + 1];
      c = __builtin_amdgcn_wmma_f32_16x16x4_f32(
          /*neg_a=*/false, a, /*neg_b=*/false, bone,
          /*c_mod=*/(short)0, c, /*reuse_a=*/false, /*reuse_b=*/false);
    }
    // D layout: lane L, VGPR r -> row M = (L>>4)*8 + r, col N = L&15.
    // Every column holds the same row sum; lanes 0 and 16 write the outputs.
    if ((t & 15) == 0) {
      const float prior = ws[64];
      const float scale = 50.0f / (float)S;  // mean over S, then /(2*0.1^2)
      const long rbase = (long)blockIdx.x * 16 + half * 8;
#pragma unroll
      for (int r = 0; r < 8; ++r) out[rbase + r] = c[r] * scale + prior;
    }
  }
}

// ---------------------------------------------------------------------------
extern "C" void kernel_launch(void* const* d_in, const int* in_sizes, int n_in,
                              void* d_out, int out_size, void* d_ws,
                              size_t ws_size, hipStream_t stream) {
  const float* pred = (const float*)d_in[0];
  const float* sigma = (const float*)d_in[1];
  const float* yReal = (const float*)d_in[2];
  const float* ySim = (const float*)d_in[3];
  float* out = (float*)d_out;
  float* ws = (float*)d_ws;  // ws[0..63] = prior partials, ws[64] = prior

  const int B = in_sizes[0] / 7;     // 8192
  const int S = in_sizes[2] / B;     // 4096

  prior_partial_kernel<<<(B + 127) / 128, 128, 0, stream>>>(pred, sigma, ws, B);
  prior_finalize_kernel<<<1, 64, 0, stream>>>(ws, B);
  posterior_kernel<<<B / 16, 256, 0, stream>>>(yReal, ySim, ws, out, S);
}